// Model_66219805769844
// MI455X (gfx1250) — compile-verified
//
#include <hip/hip_runtime.h>
#include <hip/hip_bf16.h>
#include <stddef.h>
#include <stdint.h>

// ---------------------------------------------------------------------------
// GraphSAGE 2-layer forward for MI455X (gfx1250, wave32)
//   layer: out = act( A_self @ W_self + mean_agg(A) @ W_neigh + b )
// Aggregation: CSR built once per call (int atomics only), then atomic-free
//              wave-per-node gather-mean (L2-resident feature reads).
// GEMM: V_WMMA_F32_16X16X4_F32; weights staged to LDS by the Tensor Data
//       Mover (TENSOR_LOAD_TO_LDS + s_wait_tensorcnt); K fully unrolled.
// ---------------------------------------------------------------------------

#define F_DIM 128          // feature width of x and h1 (both aggregations 128-wide)
#define K_DIM 128          // GEMM reduction depth (both layers)
#define SCAN_BLOCK 1024

typedef float v2f  __attribute__((ext_vector_type(2)));
typedef float v8f  __attribute__((ext_vector_type(8)));
typedef unsigned int u32x4 __attribute__((ext_vector_type(4)));
typedef int  i32x8 __attribute__((ext_vector_type(8)));
typedef int  i32x4 __attribute__((ext_vector_type(4)));

// ---------------------------------------------------------------- utilities
__global__ void zero_i32_kernel(int* __restrict__ p, int n) {
    int i = blockIdx.x * blockDim.x + threadIdx.x;
    int stride = gridDim.x * blockDim.x;
    for (; i < n; i += stride) p[i] = 0;
}

__global__ void copy_i32_kernel(const int* __restrict__ a, int* __restrict__ b, int n) {
    int i = blockIdx.x * blockDim.x + threadIdx.x;
    int stride = gridDim.x * blockDim.x;
    for (; i < n; i += stride) b[i] = a[i];
}

// ------------------------------------------------------------- CSR: counting
__global__ void count_degree_kernel(const int* __restrict__ dst,
                                    int* __restrict__ cnt, int n_edges) {
    int i = blockIdx.x * blockDim.x + threadIdx.x;
    int stride = gridDim.x * blockDim.x;
    for (; i < n_edges; i += stride) atomicAdd(&cnt[dst[i]], 1);
}

// ----------------------------------------------------- CSR: exclusive scan
// Single-block chunked Hillis-Steele scan: row_ptr[i] = sum(cnt[0..i-1]),
// row_ptr[n] = total edge count.
__global__ void __launch_bounds__(SCAN_BLOCK)
exclusive_scan_kernel(const int* __restrict__ cnt, int* __restrict__ row_ptr, int n) {
    __shared__ int buf[SCAN_BLOCK];
    __shared__ int carry;
    if (threadIdx.x == 0) carry = 0;
    __syncthreads();
    for (int base = 0; base < n; base += SCAN_BLOCK) {
        int i = base + (int)threadIdx.x;
        int v = (i < n) ? cnt[i] : 0;
        buf[threadIdx.x] = v;
        __syncthreads();
        #pragma unroll
        for (int off = 1; off < SCAN_BLOCK; off <<= 1) {
            int t = (threadIdx.x >= (unsigned)off) ? buf[threadIdx.x - off] : 0;
            __syncthreads();
            buf[threadIdx.x] += t;
            __syncthreads();
        }
        int incl = buf[threadIdx.x];
        if (i < n) row_ptr[i] = carry + incl - v;     // exclusive
        int chunk_total = buf[SCAN_BLOCK - 1];
        __syncthreads();
        if (threadIdx.x == 0) carry += chunk_total;
        __syncthreads();
    }
    if (threadIdx.x == 0) row_ptr[n] = carry;
}

// ------------------------------------------------------------- CSR: filling
__global__ void fill_edges_kernel(const int* __restrict__ src,
                                  const int* __restrict__ dst,
                                  int* __restrict__ cursor,
                                  int* __restrict__ edge_src, int n_edges) {
    int i = blockIdx.x * blockDim.x + threadIdx.x;
    int stride = gridDim.x * blockDim.x;
    for (; i < n_edges; i += stride) {
        int pos = atomicAdd(&cursor[dst[i]], 1);
        edge_src[pos] = src[i];
    }
}

// --------------------------------------------------- atomic-free gather-mean
// One wave per node: 32 lanes x float4 == 128 features; per-edge loads hit L2
// (feature matrix is 51 MB, fully resident in the 192 MB L2). Single b128
// store per lane; division folded in (mean). No atomics, no zero/finalize.
__global__ void __launch_bounds__(256)
gather_mean_kernel(const float* __restrict__ feat,
                   const int*   __restrict__ row_ptr,
                   const int*   __restrict__ edge_src,
                   float* __restrict__ agg, int n_nodes) {
    int wave = (int)((blockIdx.x * blockDim.x + threadIdx.x) >> 5);
    int lane = threadIdx.x & 31;
    if (wave >= n_nodes) return;            // wave-uniform
    int beg = row_ptr[wave];
    int end = row_ptr[wave + 1];
    float4 acc = make_float4(0.f, 0.f, 0.f, 0.f);
    #pragma unroll 4
    for (int j = beg; j < end; ++j) {
        int s = edge_src[j];                // wave-uniform
        float4 v = ((const float4*)(feat + (size_t)s * F_DIM))[lane];
        acc.x += v.x; acc.y += v.y; acc.z += v.z; acc.w += v.w;
    }
    float inv = 1.0f / fmaxf((float)(end - beg), 1.0f);
    acc.x *= inv; acc.y *= inv; acc.z *= inv; acc.w *= inv;
    ((float4*)(agg + (size_t)wave * F_DIM))[lane] = acc;
}

// --------------------------------------------------------- TDM weight stage
// Issue one TENSOR_LOAD_TO_LDS for a whole [K_DIM x FOUT] fp32 weight matrix
// (2-D tile == tensor; row-major; 4-byte elements). D# packing per ISA ch.8.
// This toolchain exposes the 6-arg builtin:
//   (u32x4 g0, i32x8 g1, i32x4 g2, i32x4 g3, i32x8 extra, i32 cpol)
#if __has_builtin(__builtin_amdgcn_tensor_load_to_lds)
#define HAVE_TDM 1
template <int FOUT>
__device__ __forceinline__ void tdm_load_weight(const float* W, float* ldsDst) {
    unsigned lds_addr = (unsigned)(uintptr_t)ldsDst;       // LDS aperture: low 32 bits
    unsigned long long ga = (unsigned long long)(uintptr_t)W;
    u32x4 g0;
    g0[0] = 1u;                                            // count=1 (valid, user mode)
    g0[1] = lds_addr;                                      // lds_addr
    g0[2] = (unsigned)(ga & 0xFFFFFFFFull);                // global_addr[31:0]
    g0[3] = (unsigned)((ga >> 32) & 0x1FFFFFFull)          // global_addr[56:32]
          | 0x80000000u;                                   // type=2 ("image")
    i32x8 g1;
    g1[0] = 0x20000;                                       // data_size=2 -> 4B, no mask
    g1[1] = (int)((unsigned)(FOUT & 0xFFFF) << 16);        // tensor_dim0[15:0]
    g1[2] = (int)((unsigned)(K_DIM & 0xFFFF) << 16);       // dim0[31:16]=0 | tensor_dim1[15:0]
    g1[3] = (int)((unsigned)(FOUT & 0xFFFF) << 16);        // dim1[31:16]=0 | tile_dim0=FOUT
    g1[4] = K_DIM;                                         // tile_dim1=K, tile_dim2=0
    g1[5] = FOUT;                                          // tensor_dim0_stride[31:0]
    g1[6] = 0;                                             // stride hi / dim1_stride lo
    g1[7] = 0;
    i32x4 z4 = {0, 0, 0, 0};                               // groups 2/3: 2-D tile
    i32x8 z8 = {0, 0, 0, 0, 0, 0, 0, 0};
    __builtin_amdgcn_tensor_load_to_lds(g0, g1, z4, z4, z8, 0);
}
#else
#define HAVE_TDM 0
#endif

// ----------------------------------------------------------------- SAGE GEMM
// out[N,FOUT] = act( A0[N,128] @ W0[128,FOUT] + A1[N,128] @ W1[128,FOUT] + b )
// Weights DMA'd to LDS once per block; one wave owns one 16x16 output tile.
// FOUT=128: 8 col tiles x 1 row tile / block.  FOUT=64: 4 x 2.
// N % 32 == 0 for this problem, so wave exits are uniform (EXEC all-ones).
template <int FOUT, int DO_RELU>
__global__ void __launch_bounds__(256)
sage_gemm_kernel(const float* __restrict__ A0,
                 const float* __restrict__ A1,
                 const float* __restrict__ W0,
                 const float* __restrict__ W1,
                 const float* __restrict__ bias,
                 float* __restrict__ out,
                 int nrows) {
    __shared__ float ldsW0[K_DIM * FOUT];
    __shared__ float ldsW1[K_DIM * FOUT];

    const int tid = threadIdx.x;
#if HAVE_TDM
    if (tid < 32) {                         // wave 0 issues both DMAs (EXEC ignored)
        tdm_load_weight<FOUT>(W0, ldsW0);
        tdm_load_weight<FOUT>(W1, ldsW1);
        __builtin_amdgcn_s_wait_tensorcnt(0);
    }
#else
    {
        const float4* g0 = (const float4*)W0;
        const float4* g1 = (const float4*)W1;
        float4* l0 = (float4*)ldsW0;
        float4* l1 = (float4*)ldsW1;
        #pragma unroll
        for (int i = 0; i < (K_DIM * FOUT / 4 + 255) / 256; ++i) {
            int idx = tid + i * 256;
            if (idx < K_DIM * FOUT / 4) { l0[idx] = g0[idx]; l1[idx] = g1[idx]; }
        }
    }
#endif
    __syncthreads();

    const int lane = tid & 31;
    const int wv   = tid >> 5;                      // 0..7
    constexpr int colTiles      = FOUT >> 4;        // 8 or 4
    constexpr int rowTilesPerWG = 8 / colTiles;     // 1 or 2
    const int rowTile = blockIdx.x * rowTilesPerWG + wv / colTiles;
    const int colTile = wv % colTiles;
    const int row0 = rowTile << 4;
    const int col0 = colTile << 4;
    if (row0 >= nrows) return;                      // wave-uniform exit

    // ISA 7.12.2 fp32 16x4 A-frag: lanes 0-15 -> K={k0,k0+1}; 16-31 -> K={k0+2,k0+3}
    const int m  = lane & 15;
    const int kh = (lane >> 4) << 1;                // 0 or 2
    const int cm = col0 + m;

    const float* a0p = A0 + (size_t)(row0 + m) * K_DIM + kh;
    const float* a1p = A1 + (size_t)(row0 + m) * K_DIM + kh;
    const float* lb0 = ldsW0 + cm;                  // per-lane LDS base
    const float* lb1 = ldsW1 + cm;

    v8f acc0 = {0.f,0.f,0.f,0.f,0.f,0.f,0.f,0.f};
    v8f acc1 = {0.f,0.f,0.f,0.f,0.f,0.f,0.f,0.f};

    #pragma unroll
    for (int k0 = 0; k0 < K_DIM; k0 += 4) {
        v2f a0 = *(const v2f*)(a0p + k0);           // global_load_b64, imm offset
        v2f a1 = *(const v2f*)(a1p + k0);
        v2f b0, b1;                                  // ds_load, imm offsets
        b0.x = lb0[(k0 + kh)     * FOUT];
        b0.y = lb0[(k0 + kh + 1) * FOUT];
        b1.x = lb1[(k0 + kh)     * FOUT];
        b1.y = lb1[(k0 + kh + 1) * FOUT];
        acc0 = __builtin_amdgcn_wmma_f32_16x16x4_f32(false, a0, false, b0,
                                                     (short)0, acc0, false, false);
        acc1 = __builtin_amdgcn_wmma_f32_16x16x4_f32(false, a1, false, b1,
                                                     (short)0, acc1, false, false);
    }

    // C/D layout: VGPR v -> M = v (lanes 0-15) / v+8 (lanes 16-31), N = lane%16
    float bv = bias[cm];
    const int rbase = row0 + ((lane >> 4) << 3);
    #pragma unroll
    for (int v = 0; v < 8; ++v) {
        float r = acc0[v] + acc1[v] + bv;
        if (DO_RELU) r = fmaxf(r, 0.0f);
        out[(size_t)(rbase + v) * FOUT + cm] = r;
    }
}

// ---------------------------------------------------------------------------
extern "C" void kernel_launch(void* const* d_in, const int* in_sizes, int n_in,
                              void* d_out, int out_size, void* d_ws, size_t ws_size,
                              hipStream_t stream) {
    const float* x   = (const float*)d_in[0];
    const int*   src = (const int*)  d_in[1];
    const int*   dst = (const int*)  d_in[2];
    const float* Ws1 = (const float*)d_in[3];
    const float* Wn1 = (const float*)d_in[4];
    const float* b1  = (const float*)d_in[5];
    const float* Ws2 = (const float*)d_in[6];
    const float* Wn2 = (const float*)d_in[7];
    const float* b2  = (const float*)d_in[8];
    float* out = (float*)d_out;

    const int N = in_sizes[0] / F_DIM;   // 100000
    const int E = in_sizes[1];           // 1600000

    // workspace layout
    float* agg      = (float*)d_ws;                    // N*128 f32
    float* h1       = agg + (size_t)N * F_DIM;         // N*128 f32
    int*   cnt      = (int*)(h1 + (size_t)N * F_DIM);  // N
    int*   row_ptr  = cnt + N;                         // N+1
    int*   cursor   = row_ptr + N + 1;                 // N
    int*   edge_src = cursor + N;                      // E

    dim3 blk(256);
    dim3 gridE((unsigned)((E + 255) / 256));
    dim3 gridN((unsigned)((N + 255) / 256));
    dim3 gridWaveN((unsigned)((N + 7) / 8));             // 8 waves / block

    const int rowTiles = (N + 15) / 16;                  // 6250
    dim3 gridG1((unsigned)rowTiles);                     // FOUT=128 -> 1 row tile / WG
    dim3 gridG2((unsigned)((rowTiles + 1) / 2));         // FOUT=64  -> 2 row tiles / WG

    // ---------------- CSR build (once; graph shared by both layers) --------
    zero_i32_kernel<<<gridN, blk, 0, stream>>>(cnt, N);
    count_degree_kernel<<<gridE, blk, 0, stream>>>(dst, cnt, E);
    exclusive_scan_kernel<<<1, SCAN_BLOCK, 0, stream>>>(cnt, row_ptr, N);
    copy_i32_kernel<<<gridN, blk, 0, stream>>>(row_ptr, cursor, N);
    fill_edges_kernel<<<gridE, blk, 0, stream>>>(src, dst, cursor, edge_src, E);

    // ---------------- layer 1 ----------------
    gather_mean_kernel<<<gridWaveN, blk, 0, stream>>>(x, row_ptr, edge_src, agg, N);
    sage_gemm_kernel<128, 1><<<gridG1, blk, 0, stream>>>(x, agg, Ws1, Wn1, b1, h1, N);

    // ---------------- layer 2 ----------------
    gather_mean_kernel<<<gridWaveN, blk, 0, stream>>>(h1, row_ptr, edge_src, agg, N);
    sage_gemm_kernel<64, 0><<<gridG2, blk, 0, stream>>>(h1, agg, Ws2, Wn2, b2, out, N);
}